// ExpandHarmonics_45672682226371
// MI455X (gfx1250) — compile-verified
//
#include <hip/hip_runtime.h>

#define N_ASU    4
#define HMAX     60
#define GDIM     (2 * HMAX + 1)   // 121
#define MAX_MULT 5
#define BLK      256

// LDS staging layout (dwords)
#define OFF_HKL  0                       // 15 * BLK = 3840
#define OFF_WL   (15 * BLK)              // 5 * BLK
#define OFF_D    (20 * BLK)              // 5 * BLK
#define OFF_ID   (25 * BLK)              // 5 * BLK
#define STAGE_DW (30 * BLK)              // 7680 dwords = 30 KB

__device__ __forceinline__ unsigned gcd_u(unsigned a, unsigned b) {
    while (b) { unsigned t = a % b; a = b; b = t; }
    return a;
}

// Low 32 bits of a flat pointer to LDS == wave-relative LDS byte offset
// (flat LDS aperture: ADDR[31:0] is the LDS address; hardware adds LDS_BASE).
__device__ __forceinline__ unsigned lds_off(const void* p) {
    return (unsigned)(uintptr_t)p;
}

// CDNA5 async LDS<->memory DMA (ASYNCcnt-tracked); no VGPR round trip.
__device__ __forceinline__ void async_load_b128(unsigned lds, const void* g) {
    asm volatile("global_load_async_to_lds_b128 %0, %1, off"
                 :: "v"(lds), "v"((unsigned long long)(uintptr_t)g) : "memory");
}
__device__ __forceinline__ void async_store_b128(const void* g, unsigned lds) {
    asm volatile("global_store_async_from_lds_b128 %0, %1, off"
                 :: "v"((unsigned long long)(uintptr_t)g), "v"(lds) : "memory");
}
__device__ __forceinline__ void wait_asynccnt0() {
    asm volatile("s_wait_asynccnt 0x0" ::: "memory");
}

__global__ __launch_bounds__(BLK) void ExpandHarmonics_kernel(
    const int*   __restrict__ asu_id,   // N
    const int*   __restrict__ hkl,      // N*3
    const float* __restrict__ wl,       // N
    const float* __restrict__ dmin,     // N_ASU
    const float* __restrict__ Bm,       // N_ASU*9
    const int*   __restrict__ refl,     // N_ASU*G^3
    int*         __restrict__ out,      // dword view of d_out
    long long    seg_wl,                // dword offsets of segments
    long long    seg_d,
    long long    seg_id,
    int N)
{
    __shared__ float sB[N_ASU * 9];
    __shared__ float sD[N_ASU];
    __shared__ alignas(16) int sHKL[BLK * 3];
    __shared__ alignas(16) int sOut[STAGE_DW];

    const int t    = threadIdx.x;
    const int base = blockIdx.x * BLK;
    const int rows = min(BLK, N - base);
    // b128 path needs 16B-aligned segment bases: true iff N % 4 == 0.
    const bool full  = (rows == BLK);
    const bool segOK = ((N & 3) == 0);

    if (t < N_ASU * 9) sB[t] = Bm[t];
    if (t < N_ASU)     sD[t] = dmin[t];

    // ---- Input staging: hkl (stride-3 stream) -> LDS ----
    if (full) {
        // 768 dwords = 192 b128 transfers; direct memory->LDS DMA.
        if (t < (3 * BLK / 4))
            async_load_b128(lds_off(sHKL) + 16u * t,
                            hkl + (long long)base * 3 + 4 * t);
        wait_asynccnt0();
    } else {
        for (int idx = t; idx < rows * 3; idx += BLK)
            sHKL[idx] = __builtin_nontemporal_load(hkl + (long long)base * 3 + idx);
    }

    __syncthreads();

    if (t < rows) {
        const int i = base + t;
        const int   a = __builtin_nontemporal_load(asu_id + i);
        const float w = __builtin_nontemporal_load(wl + i);
        const int h = sHKL[3 * t + 0];
        const int k = sHKL[3 * t + 1];
        const int l = sHKL[3 * t + 2];

        const bool mask = (h | k | l) != 0;

        const unsigned ah = (unsigned)(h < 0 ? -h : h);
        const unsigned ak = (unsigned)(k < 0 ? -k : k);
        const unsigned al = (unsigned)(l < 0 ? -l : l);
        const int n  = (int)gcd_u(gcd_u(ah, ak), al);
        const int ns = n > 0 ? n : 1;

        // Exact division by gcd -> trunc == floor (matches jnp floor-div).
        const int h0 = h / ns;
        const int k0 = k / ns;
        const int l0 = l / ns;

        const float w0     = w * (float)ns;
        const float dmin_g = sD[a];
        const float* Ba    = &sB[9 * a];

        const float fh = (float)h0, fk = (float)k0, fl = (float)l0;
        const float s0 = Ba[0] * fh + Ba[1] * fk + Ba[2] * fl;
        const float s1 = Ba[3] * fh + Ba[4] * fk + Ba[5] * fl;
        const float s2 = Ba[6] * fh + Ba[7] * fk + Ba[8] * fl;

        const float nrm  = sqrtf(s0 * s0 + s1 * s1 + s2 * s2);
        const float d0   = 1.0f / fmaxf(nrm, 1e-6f);
        const float nmax = fminf(floorf(d0 / dmin_g), floorf(w0 / 0.95f));
        const float nmin = floorf(w0 / 1.25f) + 1.0f;

        const long long tblBase = (long long)a * (GDIM * GDIM * GDIM);

#pragma unroll
        for (int m = 0; m < MAX_MULT; ++m) {
            float na = nmin + (float)m;
            na = (na > nmax) ? 0.0f : na;
            const int ni = (int)na;

            int ha = h0 * ni, ka = k0 * ni, la = l0 * ni;

            const int ia = ha + HMAX, ja = ka + HMAX, kz = la + HMAX;
            const bool valid = ((unsigned)ia < GDIM) & ((unsigned)ja < GDIM) &
                               ((unsigned)kz < GDIM);
            int rid = -1;
            // ni==0 indexes (HMAX,HMAX,HMAX), pinned to -1 in the table:
            // skip the gather (EXEC-masked VMEM saves L2 traffic).
            if (valid && ni != 0) {
                const int ic = min(max(ia, 0), GDIM - 1);
                const int jc = min(max(ja, 0), GDIM - 1);
                const int kc = min(max(kz, 0), GDIM - 1);
                rid = refl[tblBase + ((long long)ic * GDIM + jc) * GDIM + kc];
            }

            const bool absent = rid < 0;
            if (absent) { ha = 0; ka = 0; la = 0; na = 0.0f; }
            const float ninv = absent ? 0.0f : 1.0f / fmaxf(na, 1e-6f);
            float da = d0 * ninv;
            float wa = w0 * ninv;

            if (!mask) { ha = 0; ka = 0; la = 0; wa = 0.0f; da = 0.0f; rid = 0; }

            // Stage block image (strides 15/5: coprime with 64 banks).
            sOut[OFF_HKL + 15 * t + 3 * m + 0] = ha;
            sOut[OFF_HKL + 15 * t + 3 * m + 1] = ka;
            sOut[OFF_HKL + 15 * t + 3 * m + 2] = la;
            sOut[OFF_WL  + 5 * t + m] = __float_as_int(wa);
            sOut[OFF_D   + 5 * t + m] = __float_as_int(da);
            sOut[OFF_ID  + 5 * t + m] = rid;
        }
    }

    __syncthreads();

    // ---- Writeback: block image -> global, LDS-direct DMA where aligned ----
    int* __restrict__ g_hkl = out + (long long)base * 15;
    int* __restrict__ g_wl  = out + seg_wl + (long long)base * 5;
    int* __restrict__ g_d   = out + seg_d  + (long long)base * 5;
    int* __restrict__ g_id  = out + seg_id + (long long)base * 5;

    if (full && segOK) {
        const unsigned b  = lds_off(sOut);
        // hkl segment: 3840 dwords = 960 b128 lane-transfers
        for (int q = t; q < 15 * BLK / 4; q += BLK)
            async_store_b128(g_hkl + 4 * q, b + 16u * q);
        // wl / d / id segments: 1280 dwords = 320 b128 each
        for (int q = t; q < 5 * BLK / 4; q += BLK) {
            async_store_b128(g_wl + 4 * q, b + 4u * OFF_WL + 16u * q);
            async_store_b128(g_d  + 4 * q, b + 4u * OFF_D  + 16u * q);
            async_store_b128(g_id + 4 * q, b + 4u * OFF_ID + 16u * q);
        }
        wait_asynccnt0();   // drain before ENDPGM (implicit wait would also cover)
    } else {
        for (int idx = t; idx < rows * 15; idx += BLK)
            __builtin_nontemporal_store(sOut[OFF_HKL + idx], g_hkl + idx);
        for (int idx = t; idx < rows * 5; idx += BLK) {
            __builtin_nontemporal_store(sOut[OFF_WL + idx], g_wl + idx);
            __builtin_nontemporal_store(sOut[OFF_D  + idx], g_d  + idx);
            __builtin_nontemporal_store(sOut[OFF_ID + idx], g_id + idx);
        }
    }
}

extern "C" void kernel_launch(void* const* d_in, const int* in_sizes, int n_in,
                              void* d_out, int out_size, void* d_ws, size_t ws_size,
                              hipStream_t stream) {
    const int*   asu_id = (const int*)  d_in[0];   // (N,1) int32
    const int*   hkl    = (const int*)  d_in[1];   // (N,3) int32
    const float* wl     = (const float*)d_in[2];   // (N,1) f32
    const float* dmin   = (const float*)d_in[3];   // (4,)  f32
    const float* Bm     = (const float*)d_in[4];   // (4,3,3) f32
    const int*   refl   = (const int*)  d_in[5];   // (4,121,121,121) int32

    const int N = in_sizes[0];

    // Output segments (dword offsets, return order):
    //   hkl_all 15N | wavelength_all 5N | d_all 5N | refl_id 5N
    const long long seg_wl = (long long)15 * N;
    const long long seg_d  = (long long)20 * N;
    const long long seg_id = (long long)25 * N;

    const int grid = (N + BLK - 1) / BLK;
    ExpandHarmonics_kernel<<<grid, BLK, 0, stream>>>(
        asu_id, hkl, wl, dmin, Bm, refl,
        (int*)d_out, seg_wl, seg_d, seg_id, N);
}